// NodeModule_16003048145652
// MI455X (gfx1250) — compile-verified
//
#include <hip/hip_runtime.h>
#include <cstddef>

typedef float v2f __attribute__((ext_vector_type(2)));
typedef float v8f __attribute__((ext_vector_type(8)));
typedef float f2  __attribute__((ext_vector_type(2)));

#define NTERMS 768
#define NV 6
#define D1 5
#define TILE 16
#define NTILE (NTERMS / TILE)            // 48
#define TILES_PER_MAT (NTILE * NTILE)    // 2304
#define SQ_TERMS ((size_t)NTERMS * NTERMS)       // 589824
#define TERMS_TOTAL (SQ_TERMS + NTERMS + 1)      // 590593
#define MAT_STRIDE (TERMS_TOTAL * 54)            // 31892022 floats per matrix output

// ---- Bernstein product / elevation coefficient helpers (constant-folded) ----
__host__ __device__ constexpr int pcount(int k) { return k <= 4 ? k + 1 : 9 - k; }
__host__ __device__ constexpr int pis(int k, int p) {
  return p < pcount(k) ? ((k > 4 ? k - 4 : 0) + p) : 0;
}
__host__ __device__ constexpr int pjs(int k, int p) {
  return p < pcount(k) ? (k - pis(k, p)) : 0;
}
__host__ __device__ constexpr float pw(int k, int p) {
  const float C4[5] = {1.f, 4.f, 6.f, 4.f, 1.f};
  const float C8[9] = {1.f, 8.f, 28.f, 56.f, 70.f, 56.f, 28.f, 8.f, 1.f};
  if (p >= pcount(k)) return 0.0f;
  int i = pis(k, p);
  int j = k - i;
  return C4[i] * C4[j] / C8[k];
}
__host__ __device__ constexpr float ecoef(int i, int k) {
  const float C4[5] = {1.f, 4.f, 6.f, 4.f, 1.f};
  const float C8[9] = {1.f, 8.f, 28.f, 56.f, 70.f, 56.f, 28.f, 8.f, 1.f};
  int j = k - i;
  return (j >= 0 && j <= 4) ? C4[i] * C4[j] / C8[k] : 0.0f;
}

// ---- Phase 1: interval bounds l (under) / u (over) -> ReLU relaxation coeffs ----
__global__ void minmax_kernel(const float* __restrict__ Pu,
                              const float* __restrict__ Po,
                              float* __restrict__ ws) {
  __shared__ float slo[256];
  __shared__ float shi[256];
  const int tid = threadIdx.x;
  float accLo = 0.f, accHi = 0.f;
  for (int t = tid; t < NTERMS; t += 256) {
    // under -> lower bound contribution
    float clo = 1.f, chi = 1.f;
#pragma unroll
    for (int v = 0; v < NV; ++v) {
      const float* p = Pu + ((size_t)t * NV + v) * D1;
      float mn = p[0], mx = p[0];
#pragma unroll
      for (int i = 1; i < D1; ++i) { mn = fminf(mn, p[i]); mx = fmaxf(mx, p[i]); }
      float c0 = clo * mn, c1 = clo * mx, c2 = chi * mn, c3 = chi * mx;
      clo = fminf(fminf(c0, c1), fminf(c2, c3));
      chi = fmaxf(fmaxf(c0, c1), fmaxf(c2, c3));
    }
    accLo += clo;
    // over -> upper bound contribution
    float olo = 1.f, ohi = 1.f;
#pragma unroll
    for (int v = 0; v < NV; ++v) {
      const float* p = Po + ((size_t)t * NV + v) * D1;
      float mn = p[0], mx = p[0];
#pragma unroll
      for (int i = 1; i < D1; ++i) { mn = fminf(mn, p[i]); mx = fmaxf(mx, p[i]); }
      float c0 = olo * mn, c1 = olo * mx, c2 = ohi * mn, c3 = ohi * mx;
      olo = fminf(fminf(c0, c1), fminf(c2, c3));
      ohi = fmaxf(fmaxf(c0, c1), fmaxf(c2, c3));
    }
    accHi += ohi;
  }
  slo[tid] = accLo;
  shi[tid] = accHi;
  __syncthreads();
  for (int s = 128; s > 0; s >>= 1) {
    if (tid < s) { slo[tid] += slo[tid + s]; shi[tid] += shi[tid + s]; }
    __syncthreads();
  }
  if (tid == 0) {
    const float l = slo[0];
    const float u = shi[0];
    const float den = (u - l) > 0.f ? (u - l) : 1.0f;
    float au, bu, cu, ao, bo, co;
    if (l >= 0.f) {
      au = 0.f; bu = 1.f; cu = 0.f;
      ao = 0.f; bo = 1.f; co = 0.f;
    } else if (u <= 0.f) {
      au = 0.f; bu = 0.f; cu = 0.f;
      ao = 0.f; bo = 0.f; co = 0.f;
    } else {
      au = 1.f / den; bu = -l / den; cu = 0.f;
      const float d2 = den * den;
      ao = u / d2; bo = -2.f * u * l / d2; co = u * l * l / d2;
    }
    ws[0] = au; ws[1] = bu; ws[2] = cu;
    ws[3] = ao; ws[4] = bo; ws[5] = co;
    ws[6] = l;  ws[7] = u;
  }
}

// ---- Phase 2: sq terms via V_WMMA_F32_16X16X4_F32, LDS-restaged coalesced NT stores ----
__global__ void __launch_bounds__(32)
sq_kernel(const float* __restrict__ Pu, const float* __restrict__ Po,
          const float* __restrict__ ws, float* __restrict__ out) {
  __shared__ float lds[TILE * TILE * 18];  // 2 vars (18 coeffs) per (t,s) pair

  const int bid = blockIdx.x;
  const int mat = (bid >= TILES_PER_MAT) ? 1 : 0;
  const int tb = bid - mat * TILES_PER_MAT;
  const int t0 = (tb / NTILE) * TILE;
  const int s0 = (tb % NTILE) * TILE;
  const float* __restrict__ in = mat ? Po : Pu;
  const float aCoef = ws[mat ? 3 : 0];
  const size_t outBase = mat ? (size_t)MAT_STRIDE : (size_t)0;

  const int lane = (int)threadIdx.x;
  const int nloc = lane & 15;
  const bool hi = lane >= 16;  // lanes 16..31 hold K=2,3 for A and B fragments

  // per-lane rows: A needs P[t0+nloc, v, :], B needs P[s0+nloc, v, :]
  float pt[NV][D1], ps[NV][D1];
#pragma unroll
  for (int v = 0; v < NV; ++v)
#pragma unroll
    for (int i = 0; i < D1; ++i) {
      pt[v][i] = in[((size_t)(t0 + nloc) * NV + v) * D1 + i];
      ps[v][i] = in[((size_t)(s0 + nloc) * NV + v) * D1 + i];
    }

#pragma unroll
  for (int vp = 0; vp < 3; ++vp) {
#pragma unroll
    for (int vv = 0; vv < 2; ++vv) {
      const int v = vp * 2 + vv;
      const float aw = (v == 0) ? aCoef : 1.0f;  // sq[:,0,:] *= a folded into A
#pragma unroll
      for (int k = 0; k < 9; ++k) {
        // D[t,s] = sum_p (a? * w_p * P[t,v,i_p]) * P[s,v,j_p], K = 4 pairs per WMMA
        v2f A, B;
        A.x = aw * (hi ? pw(k, 2) * pt[v][pis(k, 2)] : pw(k, 0) * pt[v][pis(k, 0)]);
        A.y = aw * (hi ? pw(k, 3) * pt[v][pis(k, 3)] : pw(k, 1) * pt[v][pis(k, 1)]);
        B.x = hi ? ps[v][pjs(k, 2)] : ps[v][pjs(k, 0)];
        B.y = hi ? ps[v][pjs(k, 3)] : ps[v][pjs(k, 1)];
        v8f d = {0.f, 0.f, 0.f, 0.f, 0.f, 0.f, 0.f, 0.f};
        d = __builtin_amdgcn_wmma_f32_16x16x4_f32(false, A, false, B, (short)0, d,
                                                  false, false);
        if (pcount(k) > 4) {  // k==4 has 5 pairs -> second accumulating WMMA
          v2f A2, B2;
          A2.x = aw * (hi ? pw(k, 6) * pt[v][pis(k, 6)] : pw(k, 4) * pt[v][pis(k, 4)]);
          A2.y = aw * (hi ? pw(k, 7) * pt[v][pis(k, 7)] : pw(k, 5) * pt[v][pis(k, 5)]);
          B2.x = hi ? ps[v][pjs(k, 6)] : ps[v][pjs(k, 4)];
          B2.y = hi ? ps[v][pjs(k, 7)] : ps[v][pjs(k, 5)];
          d = __builtin_amdgcn_wmma_f32_16x16x4_f32(false, A2, false, B2, (short)0, d,
                                                    false, false);
        }
        // D layout: VGPR r, lanes 0-15 -> (M=r, N=lane); lanes 16-31 -> (M=r+8, N=lane-16)
#pragma unroll
        for (int r = 0; r < 8; ++r) {
          const int M = r + (hi ? 8 : 0);
          lds[(M * TILE + nloc) * 18 + vv * 9 + k] = d[r];
        }
      }
    }
    __syncthreads();
    // Epilogue: per (t,s) pair stream 18 contiguous floats (72 B, 8B-aligned) to global
#pragma unroll
    for (int it = 0; it < 8; ++it) {
      const int pairIdx = lane + 32 * it;          // 0..255
      const int tl = pairIdx >> 4;
      const int sl = pairIdx & 15;
      const size_t term = (size_t)(t0 + tl) * NTERMS + (size_t)(s0 + sl);
      float* gp = out + outBase + term * 54 + (size_t)vp * 18;
      const float* lp = &lds[pairIdx * 18];
#pragma unroll
      for (int u = 0; u < 9; ++u) {
        f2 val = *(const f2*)(lp + 2 * u);
        __builtin_nontemporal_store(val, (f2*)(gp + 2 * u));
      }
    }
    __syncthreads();
  }
}

// ---- Phase 3: lin terms (degree elevation * b) and the constant term ----
__global__ void lin_kernel(const float* __restrict__ Pu, const float* __restrict__ Po,
                           const float* __restrict__ ws, float* __restrict__ out) {
  const int idx = blockIdx.x * blockDim.x + threadIdx.x;
  if (idx >= 2 * (NTERMS + 1)) return;
  const int mat = (idx >= NTERMS + 1) ? 1 : 0;
  const int t = idx - mat * (NTERMS + 1);
  const float* __restrict__ in = mat ? Po : Pu;
  const float b = ws[mat ? 4 : 1];
  const float c = ws[mat ? 5 : 2];
  const size_t ob = mat ? (size_t)MAT_STRIDE : (size_t)0;
  if (t < NTERMS) {
    const size_t base = ob + (SQ_TERMS + (size_t)t) * 54;
#pragma unroll
    for (int v = 0; v < NV; ++v) {
      float p[D1];
#pragma unroll
      for (int i = 0; i < D1; ++i) p[i] = in[((size_t)t * NV + v) * D1 + i];
#pragma unroll
      for (int k = 0; k < 9; ++k) {
        float val = 0.f;
#pragma unroll
        for (int i = 0; i < D1; ++i)
          if (ecoef(i, k) != 0.0f) val += p[i] * ecoef(i, k);
        if (v == 0) val *= b;
        out[base + v * 9 + k] = val;
      }
    }
  } else {
    const size_t base = ob + (SQ_TERMS + (size_t)NTERMS) * 54;
#pragma unroll
    for (int v = 0; v < NV; ++v)
#pragma unroll
      for (int k = 0; k < 9; ++k) out[base + v * 9 + k] = (v == 0) ? c : 1.0f;
  }
}

extern "C" void kernel_launch(void* const* d_in, const int* in_sizes, int n_in,
                              void* d_out, int out_size, void* d_ws, size_t ws_size,
                              hipStream_t stream) {
  (void)in_sizes; (void)n_in; (void)out_size; (void)ws_size;
  const float* Pu = (const float*)d_in[0];
  const float* Po = (const float*)d_in[1];
  float* ws = (float*)d_ws;
  float* out = (float*)d_out;

  minmax_kernel<<<1, 256, 0, stream>>>(Pu, Po, ws);
  lin_kernel<<<(2 * (NTERMS + 1) + 255) / 256, 256, 0, stream>>>(Pu, Po, ws, out);
  sq_kernel<<<2 * TILES_PER_MAT, 32, 0, stream>>>(Pu, Po, ws, out);
}